// STMambaBlock_77816217469419
// MI455X (gfx1250) — compile-verified
//
#include <hip/hip_runtime.h>
#include <math.h>

// ---------------- problem constants (from reference setup_inputs) ------------
#define DIMC     128
#define D_STATE  16
#define D_CONVK  4
#define D_INNER  256
#define DT_RANK  8
#define BATCH    2
#define TT       16
#define HH       48
#define WWID     48
#define NPOS     (HH*WWID)        // 2304
// current_epoch = 12 >= ANNEAL_END(10) -> ratio = 0.5 -> keep_k = N/2
#define KEEPK    (NPOS/2)         // 1152
#define BK       (BATCH*KEEPK)    // 2304 sequences
#define ROWS     (BK*TT)          // 36864 GEMM rows (multiple of 16)
#define EPSR     1e-5f
#define XPROJ_N  (DT_RANK + 2*D_STATE)   // 40
#define XPROJ_NP 48                      // padded to 3 WMMA tiles

typedef float v2f __attribute__((ext_vector_type(2)));
typedef float v8f __attribute__((ext_vector_type(8)));

__device__ __forceinline__ float silu_f(float x){ return x / (1.0f + expf(-x)); }

// ---------------- 1) spatial energy -----------------------------------------
// Etmp[b*T+t][n] = sqrt(sum_c x[b,t,c,n]^2)
__global__ void energy_pass1(const float* __restrict__ x_in, float* __restrict__ Etmp){
  int bt = blockIdx.x;
  int n  = blockIdx.y * blockDim.x + threadIdx.x;
  if (n >= NPOS) return;
  const float* base = x_in + (size_t)bt * DIMC * NPOS + n;
  float s = 0.f;
  #pragma unroll 4
  for (int c = 0; c < DIMC; ++c){ float v = base[(size_t)c * NPOS]; s += v*v; }
  Etmp[(size_t)bt * NPOS + n] = sqrtf(s);
}

// energy[b][n] = mean_t Etmp
__global__ void energy_pass2(const float* __restrict__ Etmp, float* __restrict__ energy){
  int b = blockIdx.x;
  int n = blockIdx.y * blockDim.x + threadIdx.x;
  if (n >= NPOS) return;
  float s = 0.f;
  #pragma unroll
  for (int t = 0; t < TT; ++t) s += Etmp[(size_t)(b*TT + t) * NPOS + n];
  energy[b*NPOS + n] = s * (1.0f / TT);
}

// ---------------- 2) stable top-k + sorted compaction ------------------------
// rank(i) = #{j: e[j]>e[i]} + #{j<i: e[j]==e[i]}  (matches jax.lax.top_k ties)
__global__ __launch_bounds__(1024) void topk_kernel(const float* __restrict__ energy,
                                                    int* __restrict__ kidx,
                                                    int* __restrict__ posmap){
  __shared__ float e[NPOS];
  __shared__ int   kept[NPOS];
  __shared__ int   pref[NPOS];
  int b = blockIdx.x;
  for (int i = threadIdx.x; i < NPOS; i += blockDim.x) e[i] = energy[b*NPOS + i];
  __syncthreads();
  for (int i = threadIdx.x; i < NPOS; i += blockDim.x){
    float ei = e[i];
    int rank = 0;
    for (int j = 0; j < NPOS; ++j){
      float ej = e[j];
      rank += (ej > ei) || ((ej == ei) && (j < i));
    }
    kept[i] = (rank < KEEPK) ? 1 : 0;
  }
  __syncthreads();
  if (threadIdx.x == 0){           // short serial exclusive scan (2304 elems)
    int run = 0;
    for (int i = 0; i < NPOS; ++i){ pref[i] = run; run += kept[i]; }
  }
  __syncthreads();
  for (int i = threadIdx.x; i < NPOS; i += blockDim.x){
    if (kept[i]) { int p = pref[i]; posmap[b*NPOS + i] = p; kidx[b*KEEPK + p] = i; }
    else posmap[b*NPOS + i] = -1;
  }
}

// ---------------- 3) gather + RMSNorm1 ---------------------------------------
// XS[(b*K+k)*T + t][c] = rmsnorm(x_in[b,t,:,n])[c] * norm1_w[c]
__global__ void gather_rmsnorm(const float* __restrict__ x_in, const int* __restrict__ kidx,
                               const float* __restrict__ w, float* __restrict__ XS){
  __shared__ float red[DIMC];
  int r  = blockIdx.x;             // (b*K+k)*T + t
  int t  = r & (TT - 1);
  int bk = r >> 4;
  int b  = bk / KEEPK, k = bk % KEEPK;
  int n  = kidx[b*KEEPK + k];
  int c  = threadIdx.x;
  float v = x_in[(((size_t)(b*TT + t)) * DIMC + c) * NPOS + n];
  red[c] = v * v;
  __syncthreads();
  for (int s = DIMC/2; s > 0; s >>= 1){ if (c < s) red[c] += red[c+s]; __syncthreads(); }
  float scale = rsqrtf(red[0] * (1.0f / DIMC) + EPSR);
  XS[(size_t)r * DIMC + c] = v * scale * w[c];
}

// ---------------- generic fp32 WMMA GEMM: C = A[M,K] * W[n_valid,K]^T --------
// One 16x16 tile per wave via V_WMMA_F32_16X16X4_F32.
// MODE: 0 = plain, 1 = +bias then exact GELU, 2 = +bias
// PAD : true only for the x_proj GEMM (N=40 padded to 48); uses a branch-free
//       clamped-row + mask-multiply so the hot loop is pure load+wmma.
template<int MODE, bool PAD>
__global__ void wmma_gemm_f32(const float* __restrict__ A, const float* __restrict__ W,
                              float* __restrict__ Cout, const float* __restrict__ bias,
                              int K, int n_valid, int ldc){
  int lane = threadIdx.x & 31;
  int wave = threadIdx.x >> 5;
  int tn   = blockIdx.y * (blockDim.x >> 5) + wave;
  int row0 = blockIdx.x << 4;
  int m    = lane & 15;
  int kh   = (lane >> 4) << 1;                 // K-pair base: 0 or 2
  int col  = tn * 16 + m;
  bool cv  = !PAD || (col < n_valid);
  float wm = cv ? 1.0f : 0.0f;
  const float* Arow = A + (size_t)(row0 + m) * K + kh;
  const float* Wrow = W + (size_t)(PAD ? (cv ? col : 0) : col) * K + kh;
  v8f acc = {0.f,0.f,0.f,0.f,0.f,0.f,0.f,0.f};
  #pragma unroll 8
  for (int k = 0; k < K; k += 4){
    v2f a  = *(const v2f*)(Arow + k);          // A[m, k+kh], A[m, k+kh+1]
    v2f bf = *(const v2f*)(Wrow + k);          // B[k+kh, col] = W[col, k+kh]
    if (PAD){ bf.x *= wm; bf.y *= wm; }        // zero out padded columns
    acc = __builtin_amdgcn_wmma_f32_16x16x4_f32(false, a, false, bf,
                                                (short)0, acc, false, false);
  }
  if (PAD && !cv) return;
  float bv = (MODE != 0) ? bias[col] : 0.0f;
  int rbase = row0 + ((lane >> 4) << 3);       // D: vgpr v -> row v + 8*(lane>=16)
  #pragma unroll
  for (int v = 0; v < 8; ++v){
    float r = acc[v] + bv;
    if (MODE == 1) r = 0.5f * r * (1.0f + erff(r * 0.70710678118654752f));
    Cout[(size_t)(rbase + v) * ldc + col] = r;
  }
}

// ---------------- 4) causal depthwise conv (D_CONV=4) + SiLU -----------------
__global__ void conv_silu(const float* __restrict__ Y1, const float* __restrict__ cw,
                          const float* __restrict__ cb, float* __restrict__ XC){
  int g = blockIdx.x * blockDim.x + threadIdx.x;
  if (g >= BK * D_INNER) return;
  int d = g & (D_INNER - 1), seq = g >> 8;
  float w0 = cw[d*4+0], w1 = cw[d*4+1], w2 = cw[d*4+2], w3 = cw[d*4+3], bb = cb[d];
  float x0 = 0.f, x1 = 0.f, x2 = 0.f;
  #pragma unroll
  for (int t = 0; t < TT; ++t){
    float x3  = Y1[(size_t)(seq*TT + t) * (2*D_INNER) + d];   // xi half of in_proj
    float acc = bb + w0*x0 + w1*x1 + w2*x2 + w3*x3;
    XC[(size_t)(seq*TT + t) * D_INNER + d] = silu_f(acc);
    x0 = x1; x1 = x2; x2 = x3;
  }
}

// ---------------- 5) dt = softplus(dt_raw @ dt_w^T + dt_b), K=8 --------------
__global__ void dt_softplus(const float* __restrict__ DBL, const float* __restrict__ dtw,
                            const float* __restrict__ dtb, float* __restrict__ DT){
  int g = blockIdx.x * blockDim.x + threadIdx.x;
  if (g >= ROWS * D_INNER) return;
  int d = g & (D_INNER - 1), row = g >> 8;
  float acc = dtb[d];
  #pragma unroll
  for (int r = 0; r < DT_RANK; ++r)
    acc += DBL[(size_t)row * XPROJ_N + r] * dtw[d*DT_RANK + r];
  DT[g] = fmaxf(acc, 0.f) + log1pf(expf(-fabsf(acc)));       // stable softplus
}

// ---------------- 6) selective scan, fused +x*D and SiLU gate ----------------
// One block per sequence; thread d owns h[16] in registers; B/C tiles in LDS.
// Writes gated y in place over XC.
__global__ __launch_bounds__(D_INNER) void selective_scan(
    const float* __restrict__ DT, const float* __restrict__ DBL,
    const float* __restrict__ Y1, const float* __restrict__ A_log,
    const float* __restrict__ Dp, float* __restrict__ XC){
  __shared__ float Bs[TT][D_STATE];
  __shared__ float Cs[TT][D_STATE];
  int seq = blockIdx.x, d = threadIdx.x;
  for (int i = d; i < TT * 2 * D_STATE; i += D_INNER){
    int t = i >> 5, j = i & 31;
    float v = DBL[(size_t)(seq*TT + t) * XPROJ_N + DT_RANK + j];
    if (j < D_STATE) Bs[t][j] = v; else Cs[t][j - D_STATE] = v;
  }
  float a[D_STATE];
  #pragma unroll
  for (int s = 0; s < D_STATE; ++s) a[s] = -expf(A_log[d*D_STATE + s]);
  float Dd = Dp[d];
  __syncthreads();
  float h[D_STATE];
  #pragma unroll
  for (int s = 0; s < D_STATE; ++s) h[s] = 0.f;
  for (int t = 0; t < TT; ++t){
    size_t rb = (size_t)(seq*TT + t);
    float dt = DT[rb * D_INNER + d];
    float x  = XC[rb * D_INNER + d];
    float z  = Y1[rb * (2*D_INNER) + D_INNER + d];   // gate half of in_proj
    float dtx = dt * x;
    float y = 0.f;
    #pragma unroll
    for (int s = 0; s < D_STATE; ++s){
      float dA = expf(dt * a[s]);
      h[s] = dA * h[s] + dtx * Bs[t][s];
      y += h[s] * Cs[t][s];
    }
    y = (y + x * Dd) * silu_f(z);
    XC[rb * D_INNER + d] = y;
  }
}

// ---------------- 7) RMSNorm over contiguous 128-wide rows -------------------
__global__ void rmsnorm_rows(const float* __restrict__ src, const float* __restrict__ w,
                             float* __restrict__ dst){
  __shared__ float red[DIMC];
  int r = blockIdx.x, c = threadIdx.x;
  float v = src[(size_t)r * DIMC + c];
  red[c] = v * v;
  __syncthreads();
  for (int s = DIMC/2; s > 0; s >>= 1){ if (c < s) red[c] += red[c+s]; __syncthreads(); }
  float scale = rsqrtf(red[0] * (1.0f / DIMC) + EPSR);
  dst[(size_t)r * DIMC + c] = v * scale * w[c];
}

// ---------------- 8) scatter-add into dense output ---------------------------
__global__ void final_add(const float* __restrict__ x_in, const float* __restrict__ U,
                          const int* __restrict__ posmap, float* __restrict__ out){
  size_t idx = (size_t)blockIdx.x * blockDim.x + threadIdx.x;
  const size_t total = (size_t)BATCH * TT * DIMC * NPOS;
  if (idx >= total) return;
  int n = (int)(idx % NPOS);
  size_t r1 = idx / NPOS;
  int c = (int)(r1 % DIMC);
  size_t r2 = r1 / DIMC;
  int t = (int)(r2 % TT);
  int b = (int)(r2 / TT);
  int pos = posmap[b*NPOS + n];
  float add = 0.f;
  if (pos >= 0) add = U[((size_t)(b*KEEPK + pos) * TT + t) * DIMC + c];
  out[idx] = x_in[idx] + add;
}

// ---------------- host-side orchestration ------------------------------------
static inline size_t alignUp(size_t x, size_t a){ return (x + a - 1) & ~(a - 1); }

extern "C" void kernel_launch(void* const* d_in, const int* in_sizes, int n_in,
                              void* d_out, int out_size, void* d_ws, size_t ws_size,
                              hipStream_t stream){
  const float* x_in   = (const float*)d_in[0];
  const float* norm1w = (const float*)d_in[1];
  const float* norm2w = (const float*)d_in[2];
  const float* inW    = (const float*)d_in[3];   // [512,128]
  const float* convW  = (const float*)d_in[4];   // [256,4]
  const float* convB  = (const float*)d_in[5];
  const float* xpW    = (const float*)d_in[6];   // [40,256]
  const float* dtW    = (const float*)d_in[7];   // [256,8]
  const float* dtB    = (const float*)d_in[8];
  const float* A_log  = (const float*)d_in[9];   // [256,16]
  const float* Dp     = (const float*)d_in[10];
  const float* outW   = (const float*)d_in[11];  // [128,256]
  const float* mixW1  = (const float*)d_in[12];
  const float* mixB1  = (const float*)d_in[13];
  const float* mixW2  = (const float*)d_in[14];
  const float* mixB2  = (const float*)d_in[15];
  float* out = (float*)d_out;

  // ---- carve workspace (~252 MB) ----
  char* p = (char*)d_ws;
  size_t used = 0;
  auto take = [&](size_t bytes) -> void* {
    void* r = p + used; used += alignUp(bytes, 256); return r;
  };
  float* Etmp   = (float*)take((size_t)BATCH*TT*NPOS*4);
  float* energy = (float*)take((size_t)BATCH*NPOS*4);
  int*   kidx   = (int*)  take((size_t)BATCH*KEEPK*4);
  int*   posmap = (int*)  take((size_t)BATCH*NPOS*4);
  float* XS     = (float*)take((size_t)ROWS*DIMC*4);        // rmsnorm'd gather
  float* Y1     = (float*)take((size_t)ROWS*2*D_INNER*4);   // in_proj out (xi|z)
  float* XC     = (float*)take((size_t)ROWS*D_INNER*4);     // conv out -> scan out (in place)
  float* DBL    = (float*)take((size_t)ROWS*XPROJ_N*4);     // x_proj out [dt|B|C]
  float* DTm    = (float*)take((size_t)ROWS*D_INNER*4);     // softplus dt
  float* O1     = (float*)take((size_t)ROWS*DIMC*4);        // out_proj
  float* H1     = (float*)take((size_t)ROWS*DIMC*4);        // rmsnorm2
  float* G      = (float*)take((size_t)ROWS*DIMC*4);        // gelu(mix1)
  float* U      = (float*)take((size_t)ROWS*DIMC*4);        // mix2
  if (used > ws_size) return;  // workspace too small; bail safely

  const int NB = (NPOS + 255) / 256;  // 9

  // 1) energy
  energy_pass1<<<dim3(BATCH*TT, NB), 256, 0, stream>>>(x_in, Etmp);
  energy_pass2<<<dim3(BATCH, NB), 256, 0, stream>>>(Etmp, energy);
  // 2) top-k (stable) + sorted compaction + inverse map
  topk_kernel<<<BATCH, 1024, 0, stream>>>(energy, kidx, posmap);
  // 3) gather + rmsnorm1
  gather_rmsnorm<<<ROWS, DIMC, 0, stream>>>(x_in, kidx, norm1w, XS);
  // 4) in_proj: [ROWS,128] x [512,128]^T -> Y1 [ROWS,512]   (WMMA f32)
  wmma_gemm_f32<0,false><<<dim3(ROWS/16, 4), 256, 0, stream>>>(XS, inW, Y1, nullptr,
                                                               DIMC, 512, 512);
  // 5) depthwise conv + silu -> XC [ROWS,256]
  conv_silu<<<(BK*D_INNER + 255)/256, 256, 0, stream>>>(Y1, convW, convB, XC);
  // 6) x_proj: [ROWS,256] x [40,256]^T -> DBL [ROWS,40]  (cols padded to 48)
  wmma_gemm_f32<0,true><<<dim3(ROWS/16, 1), 96, 0, stream>>>(XC, xpW, DBL, nullptr,
                                                             D_INNER, XPROJ_N, XPROJ_N);
  // 7) dt projection + softplus
  dt_softplus<<<(ROWS*D_INNER + 255)/256, 256, 0, stream>>>(DBL, dtW, dtB, DTm);
  // 8) selective scan (fused D-residual + silu gate), in place over XC
  selective_scan<<<BK, D_INNER, 0, stream>>>(DTm, DBL, Y1, A_log, Dp, XC);
  // 9) out_proj: [ROWS,256] x [128,256]^T -> O1 [ROWS,128]
  wmma_gemm_f32<0,false><<<dim3(ROWS/16, 1), 256, 0, stream>>>(XC, outW, O1, nullptr,
                                                               D_INNER, DIMC, DIMC);
  // 10) rmsnorm2
  rmsnorm_rows<<<ROWS, DIMC, 0, stream>>>(O1, norm2w, H1);
  // 11) mix1 + bias + exact gelu
  wmma_gemm_f32<1,false><<<dim3(ROWS/16, 1), 256, 0, stream>>>(H1, mixW1, G, mixB1,
                                                               DIMC, DIMC, DIMC);
  // 12) mix2 + bias
  wmma_gemm_f32<2,false><<<dim3(ROWS/16, 1), 256, 0, stream>>>(G, mixW2, U, mixB2,
                                                               DIMC, DIMC, DIMC);
  // 13) scatter-add to dense output
  {
    size_t total = (size_t)BATCH * TT * DIMC * NPOS;
    final_add<<<(unsigned)((total + 255)/256), 256, 0, stream>>>(x_in, U, posmap, out);
  }
}